// NGCF_29729763623688
// MI455X (gfx1250) — compile-verified
//
#include <hip/hip_runtime.h>

#define NUM_USER 100000
#define NUM_ITEM 200000
#define NN (NUM_USER + NUM_ITEM)   // 300000
#define DD 64
#define LL 3
#define NNZT 5000000
#define BATCH 4096

typedef __attribute__((ext_vector_type(2))) float v2f;
typedef __attribute__((ext_vector_type(8))) float v8f;

// Native f32 global atomic add (global_atomic_add_f32, no CAS loop).
__device__ __forceinline__ void atomic_fadd(float* p, float v) {
#if defined(__has_builtin) && __has_builtin(__builtin_amdgcn_global_atomic_fadd_f32)
    __builtin_amdgcn_global_atomic_fadd_f32(p, v);
#else
    unsafeAtomicAdd(p, v);
#endif
}

// E0 = concat(user_emb, item_emb), float4 per thread
__global__ void ngcf_concat_kernel(const float* __restrict__ u,
                                   const float* __restrict__ it,
                                   float* __restrict__ E) {
    long tid  = (long)blockIdx.x * blockDim.x + threadIdx.x;
    long base = tid * 4;
    if (base >= (long)NN * DD) return;
    float4 v;
    if (base < (long)NUM_USER * DD) v = *(const float4*)(u + base);
    else                            v = *(const float4*)(it + (base - (long)NUM_USER * DD));
    *(float4*)(E + base) = v;
}

// dst = src (float4)
__global__ void ngcf_copy4_kernel(const float* __restrict__ src, float* __restrict__ dst) {
    long tid  = (long)blockIdx.x * blockDim.x + threadIdx.x;
    long base = tid * 4;
    if (base >= (long)NN * DD) return;
    *(float4*)(dst + base) = *(const float4*)(src + base);
}

// Fused SpMM: G[row] += val * (E[col] + E[col]^2)   (G pre-initialized to E)
// 16 threads per edge, each handles 4 contiguous floats.
// E (76.8 MB) is L2-resident (192 MB L2), so gathers and atomics stay on-chip.
__global__ void ngcf_spmm_kernel(const int* __restrict__ rows,
                                 const int* __restrict__ cols,
                                 const float* __restrict__ vals,
                                 const float* __restrict__ E,
                                 float* __restrict__ G) {
    long tid = (long)blockIdx.x * blockDim.x + threadIdx.x;
    long e   = tid >> 4;
    int  sub = (int)(tid & 15);
    if (e >= NNZT) return;
    int   r = rows[e];
    int   c = cols[e];
    float v = vals[e];
    const float4 x = *(const float4*)(E + (long)c * DD + sub * 4);
    float* g = G + (long)r * DD + sub * 4;
    atomic_fadd(g + 0, v * (x.x + x.x * x.x));
    atomic_fadd(g + 1, v * (x.y + x.y * x.y));
    atomic_fadd(g + 2, v * (x.z + x.z * x.z));
    atomic_fadd(g + 3, v * (x.w + x.w * x.w));
}

// In-place GE = leaky_relu(GE @ W + 2*b, 0.2), one wave per 16-row strip.
// f32 WMMA 16x16x4; W staged in LDS.
__global__ void __launch_bounds__(256)
ngcf_gemm_wmma_kernel(float* GE,                       // read A strip, write same rows
                      const float* __restrict__ W,     // 64x64 row-major
                      const float* __restrict__ bias,  // 64
                      int nStrips) {
    __shared__ float Wlds[DD * DD];
    for (int i = threadIdx.x; i < DD * DD / 4; i += blockDim.x)
        *(float4*)(Wlds + i * 4) = *(const float4*)(W + i * 4);
    __syncthreads();

    int wave  = threadIdx.x >> 5;
    int lane  = threadIdx.x & 31;
    int strip = blockIdx.x * 8 + wave;
    if (strip >= nStrips) return;

    int  half = lane >> 4;       // 0/1: which K pair / which row half
    int  mr   = lane & 15;
    long row0 = (long)strip * 16;

    // A: 16x64 strip, lane holds (M=mr, K=4c+2*half .. +1) in a[c]
    const float* Ga = GE + (row0 + mr) * DD + 2 * half;
    v2f a[16];
#pragma unroll
    for (int c = 0; c < 16; ++c)
        a[c] = *(const v2f*)(Ga + 4 * c);

#pragma unroll
    for (int t = 0; t < 4; ++t) {
        int n0 = t * 16;
        int n  = n0 + mr;
        v8f acc = {0.f, 0.f, 0.f, 0.f, 0.f, 0.f, 0.f, 0.f};
#pragma unroll
        for (int c = 0; c < 16; ++c) {
            int k = 4 * c + 2 * half;
            v2f bv;
            bv.x = Wlds[k * DD + n];
            bv.y = Wlds[(k + 1) * DD + n];
            acc = __builtin_amdgcn_wmma_f32_16x16x4_f32(
                /*neg_a=*/false, a[c], /*neg_b=*/false, bv,
                /*c_mod=*/(short)0, acc, /*reuse_a=*/false, /*reuse_b=*/false);
        }
        float bn = 2.0f * bias[n];
#pragma unroll
        for (int v = 0; v < 8; ++v) {
            float val = acc[v] + bn;
            val = fmaxf(val, 0.2f * val);          // leaky_relu(0.2)
            GE[(row0 + 8 * half + v) * DD + n] = val;
        }
    }
}

// Row L2-normalize in place; one wave per row, 2 floats per lane.
__global__ void ngcf_norm_kernel(float* __restrict__ E, int nRows) {
    int wave = threadIdx.x >> 5;
    int lane = threadIdx.x & 31;
    int row  = blockIdx.x * 8 + wave;
    if (row >= nRows) return;
    float* p = E + (long)row * DD + lane * 2;
    v2f x = *(v2f*)p;
    float s = x.x * x.x + x.y * x.y;
#pragma unroll
    for (int off = 16; off > 0; off >>= 1)
        s += __shfl_xor(s, off, 32);
    float inv = 1.0f / fmaxf(sqrtf(s), 1e-12f);
    x.x *= inv; x.y *= inv;
    *(v2f*)p = x;
}

// Scatter layer slice into output: out[g][j][colbase + d] = E[node][d]
__global__ void ngcf_gather_kernel(const float* __restrict__ E,
                                   const int* __restrict__ users,
                                   const int* __restrict__ pos,
                                   const int* __restrict__ neg,
                                   float* __restrict__ out, int colbase) {
    int tid = blockIdx.x * blockDim.x + threadIdx.x;   // float4 granularity
    if (tid >= 3 * BATCH * (DD / 4)) return;
    int d4 = tid & 15;
    int j  = (tid >> 4) & (BATCH - 1);
    int g  = tid >> 16;                                 // 4096*16 == 65536
    int node;
    if (g == 0)      node = users[j];
    else if (g == 1) node = NUM_USER + pos[j];
    else             node = NUM_USER + neg[j];
    float4 v = *(const float4*)(E + (long)node * DD + d4 * 4);
    *(float4*)(out + (long)g * BATCH * 256 + (long)j * 256 + colbase + d4 * 4) = v;
}

extern "C" void kernel_launch(void* const* d_in, const int* in_sizes, int n_in,
                              void* d_out, int out_size, void* d_ws, size_t ws_size,
                              hipStream_t stream) {
    const float* user_emb = (const float*)d_in[0];
    const float* item_emb = (const float*)d_in[1];
    const float* W        = (const float*)d_in[2];   // [3,64,64]
    const float* b        = (const float*)d_in[3];   // [3,64]
    const float* vals     = (const float*)d_in[4];
    const int*   rows     = (const int*)d_in[5];
    const int*   cols     = (const int*)d_in[6];
    const int*   users    = (const int*)d_in[7];
    const int*   pos      = (const int*)d_in[8];
    const int*   neg      = (const int*)d_in[9];
    float* out = (float*)d_out;

    float* bufA = (float*)d_ws;                      // N x 64
    float* bufB = bufA + (size_t)NN * DD;            // N x 64

    const long elems4     = (long)NN * DD / 4;       // 4.8M float4s
    const long spmmThr    = (long)NNZT * 16;
    const int  nStrips    = NN / 16;                 // 18750, exact
    const int  gatherThr  = 3 * BATCH * (DD / 4);    // 196608
    dim3 blk(256);

    ngcf_concat_kernel<<<dim3((elems4 + 255) / 256), blk, 0, stream>>>(user_emb, item_emb, bufA);
    ngcf_gather_kernel<<<dim3((gatherThr + 255) / 256), blk, 0, stream>>>(bufA, users, pos, neg, out, 0);

    float* Ecur = bufA;
    float* Enxt = bufB;
    for (int l = 0; l < LL; ++l) {
        // G = E  (then atomics add lap + sec contributions)
        ngcf_copy4_kernel<<<dim3((elems4 + 255) / 256), blk, 0, stream>>>(Ecur, Enxt);
        ngcf_spmm_kernel<<<dim3((spmmThr + 255) / 256), blk, 0, stream>>>(rows, cols, vals, Ecur, Enxt);
        // E_next = leaky_relu(G @ W_l + 2 b_l), in place
        ngcf_gemm_wmma_kernel<<<dim3((nStrips + 7) / 8), blk, 0, stream>>>(
            Enxt, W + l * DD * DD, b + l * DD, nStrips);
        ngcf_norm_kernel<<<dim3((NN + 7) / 8), blk, 0, stream>>>(Enxt, NN);
        ngcf_gather_kernel<<<dim3((gatherThr + 255) / 256), blk, 0, stream>>>(
            Enxt, users, pos, neg, out, 64 * (l + 1));
        float* tmp = Ecur; Ecur = Enxt; Enxt = tmp;
    }
}